// TrainiumBasedLinearAttention_26809185861990
// MI455X (gfx1250) — compile-verified
//
#include <hip/hip_runtime.h>

// ---------------- problem constants ----------------
#define B_    2
#define H_    8
#define BH_   16          // b*h
#define T_    1024
#define F_    16          // raw feature dim
#define FP_   160         // padded Taylor feature dim (153 -> 160)
#define HD_   64          // head dim
#define D_    512
#define CH_   16          // chunk length (WMMA tile M)
#define P_    16          // segments per sequence
#define SEG_  64          // timesteps per segment (SEG_/CH_ = 4 serial chunks)
#define SFULL (FP_ * HD_) // 10240 floats of state per (b,h)

typedef float v2f __attribute__((ext_vector_type(2)));
typedef float v8f __attribute__((ext_vector_type(8)));

// V_WMMA_F32_16X16X4_F32 : D(16x16,f32) = A(16x4,f32) * B(4x16,f32) + C
__device__ __forceinline__ v8f wmma4(v2f a, v2f b, v8f c) {
  return __builtin_amdgcn_wmma_f32_16x16x4_f32(false, a, false, b, (short)0, c,
                                               false, false);
}

// 32-bit LDS byte offset of a __shared__ object (generic -> addrspace(3))
__device__ __forceinline__ unsigned lds_off(const void* p) {
  return (unsigned)(unsigned long long)(__attribute__((address_space(3))) const
                                        void*)p;
}

// GLOBAL_LOAD_ASYNC_TO_LDS_B128: 16 B/lane, tracked by ASYNCcnt
__device__ __forceinline__ void async_ld_b128(unsigned ldsaddr,
                                              const void* gaddr) {
  asm volatile("global_load_async_to_lds_b128 %0, %1, off" ::"v"(ldsaddr),
               "v"(gaddr)
               : "memory");
}
__device__ __forceinline__ void wait_async0() {
  asm volatile("s_wait_asynccnt 0x0" ::: "memory");
}

// ============================================================
// Kernel 1: fused QKV projection.  X(2048x512) @ {Wq,Wk,Wv}^T
// one wave computes one 16x16 output tile, K-loop of 128 WMMAs.
// Output is scattered into (b,h,t,·) layout for the scan.
// ============================================================
__global__ __launch_bounds__(32) void gemm_qkv_kernel(
    const float* __restrict__ X, const float* __restrict__ Wq,
    const float* __restrict__ Wk, const float* __restrict__ Wv,
    float* __restrict__ qraw, float* __restrict__ kraw,
    float* __restrict__ vbuf) {
  const int mt   = blockIdx.x;       // 128 row tiles
  const int n0   = blockIdx.y * 16;  // 0..767 (128 q | 128 k | 512 v)
  const int lane = threadIdx.x & 31;
  const int col  = lane & 15;
  const int koff = (lane >> 4) * 2;

  const float* W;
  int wr0;
  if (n0 < 128)      { W = Wq; wr0 = n0; }
  else if (n0 < 256) { W = Wk; wr0 = n0 - 128; }
  else               { W = Wv; wr0 = n0 - 256; }

  const float* arow = X + (size_t)(mt * 16 + col) * D_;
  const float* brow = W + (size_t)(wr0 + col) * D_;

  v8f acc = {};
  for (int k = 0; k < D_; k += 4) {
    v2f a = *(const v2f*)(arow + k + koff);
    v2f b = *(const v2f*)(brow + k + koff);
    acc = wmma4(a, b, acc);
  }

#pragma unroll
  for (int r = 0; r < 8; ++r) {
    int M  = mt * 16 + ((lane < 16) ? r : r + 8);
    int bb = M >> 10, t = M & (T_ - 1);
    int n  = n0 + col;
    float val = acc[r];
    if (n < 128) {
      int hh = n >> 4, f = n & 15;
      qraw[((size_t)(bb * H_ + hh) * T_ + t) * F_ + f] = val;
    } else if (n < 256) {
      int nn = n - 128, hh = nn >> 4, f = nn & 15;
      kraw[((size_t)(bb * H_ + hh) * T_ + t) * F_ + f] = val;
    } else {
      int nn = n - 256, hh = nn >> 6, dd = nn & 63;
      vbuf[((size_t)(bb * H_ + hh) * T_ + t) * HD_ + dd] = val;
    }
  }
}

// ============================================================
// Kernel 2: Taylor feature map.  16 -> 160 (153 real + 7 zero pad)
//   [1, x/d^{1/4}, x^2/(sqrt(d)*sqrt2), x_i x_j / sqrt(d) (i<j)]
//   d = 16 : d^{1/4}=2, sqrt(d)=4
// ============================================================
__global__ __launch_bounds__(256) void featmap_kernel(
    const float* __restrict__ qraw, const float* __restrict__ kraw,
    float* __restrict__ Qf, float* __restrict__ Kf) {
  const int rows = BH_ * T_;
  int idx = blockIdx.x * 256 + threadIdx.x;
  if (idx >= 2 * rows) return;
  const float* src = (idx < rows) ? (qraw + (size_t)idx * F_)
                                  : (kraw + (size_t)(idx - rows) * F_);
  float* dst = (idx < rows) ? (Qf + (size_t)idx * FP_)
                            : (Kf + (size_t)(idx - rows) * FP_);
  float x[F_];
#pragma unroll
  for (int i = 0; i < F_; ++i) x[i] = src[i];
  dst[0] = 1.0f;
#pragma unroll
  for (int i = 0; i < F_; ++i) dst[1 + i] = x[i] * 0.5f;
#pragma unroll
  for (int i = 0; i < F_; ++i)
    dst[17 + i] = x[i] * x[i] * 0.17677669529663687f;  // 1/(4*sqrt2)
  int p = 33;
  for (int i = 0; i < F_; ++i)
    for (int j = i + 1; j < F_; ++j) dst[p++] = x[i] * x[j] * 0.25f;
  for (int q = 153; q < FP_; ++q) dst[q] = 0.0f;
}

// ============================================================
// Kernel 3 (pass 1): per-segment state  S_seg = K_seg^T @ V_seg
// (160x64, K=64) + per-segment ksum.  256 blocks, 8 waves each.
// K/V segment staged to LDS once via async b128 copies, then
// 40 WMMA tile-jobs (10 Mtiles x 4 Ntiles) read LDS only.
// ============================================================
__global__ __launch_bounds__(256) void seg_state_kernel(
    const float* __restrict__ Kf, const float* __restrict__ V,
    float* __restrict__ Sseg, float* __restrict__ kseg) {
  __shared__ __align__(16) float Ks[SEG_ * FP_];  // 40 KB
  __shared__ __align__(16) float Vs[SEG_ * HD_];  // 16 KB

  const int bh = blockIdx.x, seg = blockIdx.y;
  const int tid = threadIdx.x, wave = tid >> 5, lane = tid & 31;
  const int col = lane & 15, koff = (lane >> 4) * 2;
  const float* Kb = Kf + (size_t)bh * T_ * FP_;
  const float* Vb = V + (size_t)bh * T_ * HD_;
  const int ts0 = seg * SEG_;
  float* Sout = Sseg + (size_t)(bh * P_ + seg) * SFULL;

  // ---- async-stage the whole segment: 80 x 512B (K) + 32 x 512B (V) ----
  {
    const char* gk = (const char*)(Kb + (size_t)ts0 * FP_);
    const char* gv = (const char*)(Vb + (size_t)ts0 * HD_);
    const unsigned lk = lds_off(Ks), lv = lds_off(Vs);
    for (int j = wave; j < 112; j += 8) {
      if (j < 80) {
        int boff = j * 512 + lane * 16;
        async_ld_b128(lk + boff, gk + boff);
      } else {
        int boff = (j - 80) * 512 + lane * 16;
        async_ld_b128(lv + boff, gv + boff);
      }
    }
    wait_async0();
  }
  __syncthreads();

  for (int job = wave; job < 40; job += 8) {
    int mt = job >> 2, nt = job & 3;
    v8f cc = {};
    for (int k = 0; k < SEG_; k += 4) {
      v2f a, b;
      a.x = Ks[(k + koff) * FP_ + mt * 16 + col];
      a.y = Ks[(k + koff + 1) * FP_ + mt * 16 + col];
      const float* vp = &Vs[(k + koff) * HD_ + nt * 16 + col];
      b.x = vp[0];
      b.y = vp[HD_];
      cc = wmma4(a, b, cc);
    }
#pragma unroll
    for (int r = 0; r < 8; ++r) {
      int M = mt * 16 + ((lane < 16) ? r : r + 8);
      Sout[(size_t)M * HD_ + nt * 16 + col] = cc[r];
    }
  }
  for (int f = tid; f < FP_; f += 256) {
    float s = 0.f;
    for (int j = 0; j < SEG_; ++j) s += Ks[j * FP_ + f];
    kseg[(size_t)(bh * P_ + seg) * FP_ + f] = s;
  }
}

// ============================================================
// Kernel 4 (pass 2): exclusive prefix over 16 segments
// (per (b,h), elementwise over the 10240-elem state + 160 ksum)
// ============================================================
__global__ __launch_bounds__(256) void prefix_kernel(
    const float* __restrict__ Sseg, float* __restrict__ Spre,
    const float* __restrict__ kseg, float* __restrict__ kpre) {
  const int NS = BH_ * SFULL;
  const int NK = BH_ * FP_;
  int idx = blockIdx.x * 256 + threadIdx.x;
  if (idx < NS) {
    int bh = idx / SFULL, e = idx % SFULL;
    float run = 0.f;
    for (int s = 0; s < P_; ++s) {
      Spre[(size_t)(bh * P_ + s) * SFULL + e] = run;
      run += Sseg[(size_t)(bh * P_ + s) * SFULL + e];
    }
  } else if (idx < NS + NK) {
    int j = idx - NS;
    int bh = j / FP_, f = j % FP_;
    float run = 0.f;
    for (int s = 0; s < P_; ++s) {
      kpre[(size_t)(bh * P_ + s) * FP_ + f] = run;
      run += kseg[(size_t)(bh * P_ + s) * FP_ + f];
    }
  }
}

// ============================================================
// Kernel 5 (pass 3): intra-segment chunked causal linear attention.
// 256 blocks (bh x seg), 8 waves.  S (160x64 fp32) lives in LDS.
// Per 16-row chunk: Q/K/V chunk async-staged to LDS once, then
//   phase1: waves0-3  acc  = Qc @ S           (4 N-tiles, K=160)
//           wave4     A    = mask(Qc @ Kc^T)  -> LDS
//           wave5     qdot = Qc . ksum        (den inter term)
//           waves6-7  kdelta = colsum(Kc)
//   phase2: waves0-3  acc += mask(A) @ Vc     (K=16)
//           wave4     den  = qdot + rowsum(A) + eps
//           waves5-7  ksum += kdelta
//   phase3: waves0-3  Y = acc / den ; all 8 waves: S += Kc^T @ Vc
// ============================================================
__global__ __launch_bounds__(256) void scan_kernel(
    const float* __restrict__ Qf, const float* __restrict__ Kf,
    const float* __restrict__ V, const float* __restrict__ Spre,
    const float* __restrict__ kpre, float* __restrict__ Y) {
  __shared__ float S_lds[SFULL];                    // 40 KB state
  __shared__ __align__(16) float Q_lds[CH_ * FP_];  // 10 KB chunk stage
  __shared__ __align__(16) float K_lds[CH_ * FP_];  // 10 KB
  __shared__ __align__(16) float V_lds[CH_ * HD_];  // 4 KB
  __shared__ float A_lds[CH_ * CH_];
  __shared__ float ksum[FP_];
  __shared__ float kdelta[FP_];
  __shared__ float den_lds[CH_];
  __shared__ float qdot_lds[CH_];

  const int bh = blockIdx.x, seg = blockIdx.y;
  const int tid = threadIdx.x, wave = tid >> 5, lane = tid & 31;
  const int col = lane & 15, koff = (lane >> 4) * 2;

  const float* Qb = Qf + (size_t)bh * T_ * FP_;
  const float* Kb = Kf + (size_t)bh * T_ * FP_;
  const float* Vb = V + (size_t)bh * T_ * HD_;
  const int bb = bh >> 3, hh = bh & 7;
  float* Yb = Y + (size_t)bb * T_ * D_ + hh * HD_;

  const float* Sp = Spre + (size_t)(bh * P_ + seg) * SFULL;
  const float* Kp = kpre + (size_t)(bh * P_ + seg) * FP_;
  for (int i = tid; i < SFULL; i += 256) S_lds[i] = Sp[i];
  for (int i = tid; i < FP_; i += 256) ksum[i] = Kp[i];
  __syncthreads();

  const unsigned lq = lds_off(Q_lds), lk = lds_off(K_lds), lv = lds_off(V_lds);

  for (int c = 0; c < SEG_ / CH_; ++c) {
    const int t0 = seg * SEG_ + c * CH_;
    v8f acc = {};

    // ---- async-stage Qc/Kc/Vc (each contiguous in global) ----
    {
      const char* gq = (const char*)(Qb + (size_t)t0 * FP_);
      const char* gk = (const char*)(Kb + (size_t)t0 * FP_);
      const char* gv = (const char*)(Vb + (size_t)t0 * HD_);
      for (int j = wave; j < 48; j += 8) {  // 20 Q + 20 K + 8 V x 512B
        if (j < 20) {
          int boff = j * 512 + lane * 16;
          async_ld_b128(lq + boff, gq + boff);
        } else if (j < 40) {
          int boff = (j - 20) * 512 + lane * 16;
          async_ld_b128(lk + boff, gk + boff);
        } else {
          int boff = (j - 40) * 512 + lane * 16;
          async_ld_b128(lv + boff, gv + boff);
        }
      }
      wait_async0();
    }
    __syncthreads();

    // ---------- phase 1 ----------
    if (wave < 4) {
      const float* qrow = &Q_lds[col * FP_];
      for (int k = 0; k < FP_; k += 4) {
        v2f a = *(const v2f*)(qrow + k + koff);
        v2f b;
        b.x = S_lds[(k + koff) * HD_ + wave * 16 + col];
        b.y = S_lds[(k + koff + 1) * HD_ + wave * 16 + col];
        acc = wmma4(a, b, acc);
      }
    } else if (wave == 4) {
      v8f at = {};
      const float* qrow = &Q_lds[col * FP_];
      const float* krow = &K_lds[col * FP_];
      for (int k = 0; k < FP_; k += 4) {
        v2f a = *(const v2f*)(qrow + k + koff);
        v2f b = *(const v2f*)(krow + k + koff);
        at = wmma4(a, b, at);
      }
#pragma unroll
      for (int r = 0; r < 8; ++r) {
        int M = (lane < 16) ? r : r + 8;
        A_lds[M * CH_ + col] = (col <= M) ? at[r] : 0.0f;  // causal, inclusive
      }
    } else if (wave == 5) {
      if (lane < CH_) {
        const float* qrow = &Q_lds[lane * FP_];
        float s = 0.f;
        for (int f = 0; f < FP_; ++f) s += qrow[f] * ksum[f];
        qdot_lds[lane] = s;
      }
    } else {
      for (int f = (wave - 6) * 32 + lane; f < FP_; f += 64) {
        float s = 0.f;
        for (int j = 0; j < CH_; ++j) s += K_lds[j * FP_ + f];
        kdelta[f] = s;
      }
    }
    __syncthreads();

    // ---------- phase 2 ----------
    if (wave < 4) {
      for (int k = 0; k < CH_; k += 4) {
        v2f a = *(const v2f*)(&A_lds[col * CH_ + k + koff]);
        const float* vp = &V_lds[(k + koff) * HD_ + wave * 16 + col];
        v2f b;
        b.x = vp[0];
        b.y = vp[HD_];
        acc = wmma4(a, b, acc);
      }
    } else if (wave == 4) {
      if (lane < CH_) {
        float s = 0.f;
        for (int j = 0; j <= lane; ++j) s += A_lds[lane * CH_ + j];
        den_lds[lane] = qdot_lds[lane] + s + 1e-12f;
      }
    } else {
      for (int f = (wave - 5) * 32 + lane; f < FP_; f += 96) ksum[f] += kdelta[f];
    }
    __syncthreads();

    // ---------- phase 3 ----------
    if (wave < 4) {
#pragma unroll
      for (int r = 0; r < 8; ++r) {
        int M = (lane < 16) ? r : r + 8;
        Yb[(size_t)(t0 + M) * D_ + wave * 16 + col] = acc[r] / den_lds[M];
      }
    }
    for (int job = wave; job < 40; job += 8) {  // S += Kc^T @ Vc
      int mt = job >> 2, nt = job & 3;
      v8f cc;
#pragma unroll
      for (int r = 0; r < 8; ++r) {
        int M = mt * 16 + ((lane < 16) ? r : r + 8);
        cc[r] = S_lds[M * HD_ + nt * 16 + col];
      }
      for (int k = 0; k < CH_; k += 4) {
        v2f a, b;
        a.x = K_lds[(k + koff) * FP_ + mt * 16 + col];
        a.y = K_lds[(k + koff + 1) * FP_ + mt * 16 + col];
        const float* vp = &V_lds[(k + koff) * HD_ + nt * 16 + col];
        b.x = vp[0];
        b.y = vp[HD_];
        cc = wmma4(a, b, cc);
      }
#pragma unroll
      for (int r = 0; r < 8; ++r) {
        int M = mt * 16 + ((lane < 16) ? r : r + 8);
        S_lds[M * HD_ + nt * 16 + col] = cc[r];
      }
    }
    __syncthreads();
  }
}

// ============================================================
// Kernel 6: output projection  out = Y(2048x512) @ Wo^T(512x512)
// ============================================================
__global__ __launch_bounds__(32) void gemm_out_kernel(
    const float* __restrict__ Yin, const float* __restrict__ Wo,
    float* __restrict__ Out) {
  const int mt = blockIdx.x, nt = blockIdx.y;
  const int lane = threadIdx.x & 31;
  const int col = lane & 15, koff = (lane >> 4) * 2;
  const float* arow = Yin + (size_t)(mt * 16 + col) * D_;
  const float* brow = Wo + (size_t)(nt * 16 + col) * D_;
  v8f acc = {};
  for (int k = 0; k < D_; k += 4) {
    v2f a = *(const v2f*)(arow + k + koff);
    v2f b = *(const v2f*)(brow + k + koff);
    acc = wmma4(a, b, acc);
  }
#pragma unroll
  for (int r = 0; r < 8; ++r) {
    int M = mt * 16 + ((lane < 16) ? r : r + 8);
    Out[(size_t)M * D_ + nt * 16 + col] = acc[r];
  }
}

// ============================================================
extern "C" void kernel_launch(void* const* d_in, const int* in_sizes, int n_in,
                              void* d_out, int out_size, void* d_ws,
                              size_t ws_size, hipStream_t stream) {
  const float* hidden = (const float*)d_in[0];
  const float* Wq = (const float*)d_in[1];
  const float* Wk = (const float*)d_in[2];
  const float* Wv = (const float*)d_in[3];
  const float* Wo = (const float*)d_in[4];
  float* out = (float*)d_out;
  float* ws = (float*)d_ws;

  float* qraw = ws;                                  // 16*1024*16
  float* kraw = qraw + (size_t)BH_ * T_ * F_;        // 16*1024*16
  float* vbuf = kraw + (size_t)BH_ * T_ * F_;        // 16*1024*64
  float* Qf   = vbuf + (size_t)BH_ * T_ * HD_;       // 16*1024*160
  float* Kf   = Qf + (size_t)BH_ * T_ * FP_;         // 16*1024*160
  float* ybuf = Kf + (size_t)BH_ * T_ * FP_;         // 2*1024*512
  float* Sseg = ybuf + (size_t)B_ * T_ * D_;         // 16*16*10240
  float* Spre = Sseg + (size_t)BH_ * P_ * SFULL;     // 16*16*10240
  float* kseg = Spre + (size_t)BH_ * P_ * SFULL;     // 16*16*160
  float* kpre = kseg + (size_t)BH_ * P_ * FP_;       // 16*16*160

  gemm_qkv_kernel<<<dim3(128, 48), 32, 0, stream>>>(hidden, Wq, Wk, Wv, qraw,
                                                    kraw, vbuf);
  featmap_kernel<<<dim3(128), 256, 0, stream>>>(qraw, kraw, Qf, Kf);
  seg_state_kernel<<<dim3(BH_, P_), 256, 0, stream>>>(Kf, vbuf, Sseg, kseg);
  {
    int njobs = BH_ * SFULL + BH_ * FP_;
    prefix_kernel<<<dim3((njobs + 255) / 256), 256, 0, stream>>>(Sseg, Spre,
                                                                 kseg, kpre);
  }
  scan_kernel<<<dim3(BH_, P_), 256, 0, stream>>>(Qf, Kf, vbuf, Spre, kpre,
                                                 ybuf);
  gemm_out_kernel<<<dim3(128, 32), 32, 0, stream>>>(ybuf, Wo, out);
}